// TPSSpatialTransformer_1760936591555
// MI455X (gfx1250) — compile-verified
//
#include <hip/hip_runtime.h>

typedef __attribute__((ext_vector_type(2))) float v2f;
typedef __attribute__((ext_vector_type(8))) float v8f;

#define H_OUT   128
#define W_OUT   512
#define HW      (H_OUT * W_OUT)   // 65536
#define NB      256
#define NCTRL   20
#define KDIM    23                // N_CTRL + 3
#define KPAD    24                // pad K to 6 steps of 4
#define BGROUPS 32                // 256 batches / 8 per group
#define BBPG    8                 // batches packed per 16-col WMMA tile (x,y per batch)

// ---------------------------------------------------------------------------
// Kernel 1: build padded B-operand table  bpad[32][24][16]  (48 KB in d_ws)
//   col c = 2*bb + coord,  batch b = bg*8 + bb
//   bpad[bg][k][c] = mapping[b][k][coord] = sum_{j<20} invK[k][j]*scp[b][j][coord]
//   row k=23 is zero (K padding).
// ---------------------------------------------------------------------------
__global__ __launch_bounds__(384)
void tps_mapping_kernel(const float* __restrict__ scp,
                        const float* __restrict__ invK,
                        float* __restrict__ bpad) {
    int bg = blockIdx.x;          // 0..31
    int t  = threadIdx.x;         // 0..383
    int k  = t >> 4;              // 0..23
    int c  = t & 15;
    int bb = c >> 1;
    int coord = c & 1;
    int b  = bg * BBPG + bb;
    float acc = 0.0f;
    if (k < KDIM) {
        const float* Krow = invK + k * KDIM;
        const float* S    = scp + (size_t)b * NCTRL * 2 + coord;
#pragma unroll
        for (int j = 0; j < NCTRL; ++j)
            acc = fmaf(Krow[j], S[2 * j], acc);
    }
    bpad[(size_t)bg * (KPAD * 16) + k * 16 + c] = acc;
}

// ---------------------------------------------------------------------------
// Kernel 2: source_coordinate via V_WMMA_F32_16X16X4_F32.
//   One wave = one 16-pixel tile of tcr (A, kept in registers), loops over
//   32 batch-groups (B from L2-resident bpad).  D tile = 16 px x (8 batches
//   x 2 coords).  6 WMMAs per (tile, batch-group), K padded 23->24.
// A layout (32-bit A 16x4): lane m (0..15): K pair {4k,4k+1}; lane m+16: {4k+2,4k+3}
// B layout (32-bit B 4x16): lanes 0-15 rows {4k,4k+1}; lanes 16-31 rows {4k+2,4k+3}
// D layout: VGPR g -> row g (lanes<16) / row g+8 (lanes>=16), col = lane&15
// ---------------------------------------------------------------------------
__global__ __launch_bounds__(256)
void tps_coord_wmma_kernel(const float* __restrict__ tcr,
                           const float* __restrict__ bpad,
                           float* __restrict__ sc) {
    int wave = (int)((blockIdx.x * blockDim.x + threadIdx.x) >> 5); // tile 0..4095
    int lane = threadIdx.x & 31;
    int m    = lane & 15;
    int hs   = lane >> 4;         // half-select for K pairs
    int p0   = wave * 16;

    // Load A tile: 16 pixels x 24 K (last K zero-padded), 2 floats per lane per kstep
    v2f a[6];
    const float* arow = tcr + (size_t)(p0 + m) * KDIM;
#pragma unroll
    for (int ks = 0; ks < 6; ++ks) {
        int k0 = 4 * ks + 2 * hs;
        float x = arow[k0];
        float y = (k0 + 1 < KDIM) ? arow[k0 + 1] : 0.0f;  // only ks=5,hs=1 pads
        a[ks].x = x;
        a[ks].y = y;
    }

    int c     = lane & 15;
    int bb    = c >> 1;
    int coord = c & 1;

    for (int bg = 0; bg < BGROUPS; ++bg) {
        const float* Bt = bpad + (size_t)bg * (KPAD * 16);
        v8f acc = {};
#pragma unroll
        for (int ks = 0; ks < 6; ++ks) {
            int k0 = 4 * ks + 2 * hs;
            v2f bv;
            bv.x = Bt[k0 * 16 + c];
            bv.y = Bt[(k0 + 1) * 16 + c];
            acc = __builtin_amdgcn_wmma_f32_16x16x4_f32(
                /*neg_a=*/false, a[ks], /*neg_b=*/false, bv,
                /*c_mod=*/(short)0, acc, /*reuse_a=*/false, /*reuse_b=*/false);
        }
        int b = bg * BBPG + bb;
        float* ob = sc + ((size_t)b * HW) * 2 + coord;
#pragma unroll
        for (int g = 0; g < 8; ++g) {
            int pix = p0 + g + 8 * hs;
            ob[(size_t)pix * 2] = acc[g];
        }
    }
}

// ---------------------------------------------------------------------------
// Kernel 3: bilinear grid-sample, align_corners=True semantics.
//   ix = clip(sc_x,0,1)*(W-1)  (clip->[-1,1]->unnormalize fused)
//   One thread per (b, pixel), all 3 channels (sc read once).
// ---------------------------------------------------------------------------
__global__ __launch_bounds__(256)
void tps_sample_kernel(const float* __restrict__ img,
                       const float* __restrict__ sc,
                       float* __restrict__ out) {
    size_t idx = (size_t)blockIdx.x * blockDim.x + threadIdx.x; // b*HW + p
    int b = (int)(idx >> 16);
    int p = (int)(idx & (HW - 1));

    float gx = sc[idx * 2 + 0];
    float gy = sc[idx * 2 + 1];
    gx = fminf(fmaxf(gx, 0.0f), 1.0f);
    gy = fminf(fmaxf(gy, 0.0f), 1.0f);
    float ix = gx * (float)(W_OUT - 1);
    float iy = gy * (float)(H_OUT - 1);
    float x0f = floorf(ix), y0f = floorf(iy);
    float wx = ix - x0f, wy = iy - y0f;
    int x0 = min(max((int)x0f, 0), W_OUT - 1);
    int y0 = min(max((int)y0f, 0), H_OUT - 1);
    int x1 = min(x0 + 1, W_OUT - 1);
    int y1 = min(y0 + 1, H_OUT - 1);

    float w00 = (1.0f - wx) * (1.0f - wy);
    float w01 = wx * (1.0f - wy);
    float w10 = (1.0f - wx) * wy;
    float w11 = wx * wy;

    const float* base = img + (size_t)b * 3 * HW;
#pragma unroll
    for (int ch = 0; ch < 3; ++ch) {
        const float* pl = base + (size_t)ch * HW;
        float v00 = pl[y0 * W_OUT + x0];
        float v01 = pl[y0 * W_OUT + x1];
        float v10 = pl[y1 * W_OUT + x0];
        float v11 = pl[y1 * W_OUT + x1];
        out[((size_t)(b * 3 + ch)) * HW + p] =
            v00 * w00 + v01 * w01 + v10 * w10 + v11 * w11;
    }
}

// ---------------------------------------------------------------------------
extern "C" void kernel_launch(void* const* d_in, const int* in_sizes, int n_in,
                              void* d_out, int out_size, void* d_ws, size_t ws_size,
                              hipStream_t stream) {
    const float* input = (const float*)d_in[0];  // [256,3,128,512]
    const float* scp   = (const float*)d_in[1];  // [256,20,2]
    const float* invK  = (const float*)d_in[2];  // [23,23]
    const float* tcr   = (const float*)d_in[3];  // [65536,23]

    float* out_maps = (float*)d_out;                       // [256,3,128,512]
    float* sc       = out_maps + (size_t)NB * 3 * HW;      // [256,65536,2]
    float* bpad     = (float*)d_ws;                        // 32*24*16 f32 = 48 KB

    // 1) tiny mapping GEMM -> padded WMMA B-operand table
    tps_mapping_kernel<<<BGROUPS, KPAD * 16, 0, stream>>>(scp, invK, bpad);

    // 2) coordinate GEMM via f32 WMMA: 4096 tiles, 8 waves/block -> 512 blocks
    tps_coord_wmma_kernel<<<512, 256, 0, stream>>>(tcr, bpad, sc);

    // 3) fused clip + bilinear sample: 256*65536 threads
    tps_sample_kernel<<<(NB * HW) / 256, 256, 0, stream>>>(input, sc, out_maps);
}